// MyModel_87522843558569
// MI455X (gfx1250) — compile-verified
//
#include <hip/hip_runtime.h>

typedef float v2f __attribute__((ext_vector_type(2)));
typedef float v8f __attribute__((ext_vector_type(8)));

#define T_STEPS 512
#define BATCH   64
#define DIM     1024
#define UNITS   1024
#define G4      4096
#define NBLOCKS 64          // one block per 16-wide u slice (owns all 4 gates)
#define NTHREADS 1024       // 32 waves: [khalf(2)][gate(4)][rowtile(4)]
#define RT_STRIDE 1028      // 1024 + 4 pad -> conflict-free ds_load_b64
#define SMEM_FLOATS (64 * RT_STRIDE + 2 * 64 * 65)

__global__ void lstm_init(float* cstate, unsigned* ctr) {
    int i = blockIdx.x * blockDim.x + threadIdx.x;
    if (i < BATCH * UNITS) cstate[i] = 0.0f;
    if (i == 0) *ctr = 0u;
}

__device__ __forceinline__ float sigf(float x) { return 1.0f / (1.0f + __expf(-x)); }

// ------- Phase 1: xz = x @ W + bias. Wave computes a 64x16 strip (4 row-tiles    -------
// ------- share each B-fragment): 4 b64 A-loads + 2 b32 B-loads per 4 WMMA.       -------
// ------- xz is written with NT hint: 536 MB write-once stream, don't pollute L2. -------
__global__ void __launch_bounds__(256, 4)
xz_gemm(const float* __restrict__ x, const float* __restrict__ W,
        const float* __restrict__ bias, float* __restrict__ xz)
{
    const int lane  = threadIdx.x & 31;
    const int wave  = threadIdx.x >> 5;           // 8 waves/block
    const int strip = blockIdx.x * 8 + wave;      // 131072 strips of 64x16
    const int cs    = strip & 255;                // col strip (4096/16)
    const int rm    = strip >> 8;                 // row macro (32768/64)
    const int half  = lane >> 4;
    const int n     = lane & 15;
    const int col   = cs * 16 + n;

    const float* a0p = x + (size_t)(rm * 64 +  0 + n) * DIM;
    const float* a1p = x + (size_t)(rm * 64 + 16 + n) * DIM;
    const float* a2p = x + (size_t)(rm * 64 + 32 + n) * DIM;
    const float* a3p = x + (size_t)(rm * 64 + 48 + n) * DIM;
    const float* bp  = W + col;

    v8f c0 = {}, c1 = {}, c2 = {}, c3 = {};
    for (int k = 0; k < DIM; k += 4) {
        const int ko = k + 2 * half;
        v2f b;
        b.x = bp[(size_t)ko * G4];
        b.y = bp[(size_t)(ko + 1) * G4];
        v2f a0 = *(const v2f*)(a0p + ko);
        v2f a1 = *(const v2f*)(a1p + ko);
        v2f a2 = *(const v2f*)(a2p + ko);
        v2f a3 = *(const v2f*)(a3p + ko);
        c0 = __builtin_amdgcn_wmma_f32_16x16x4_f32(false, a0, false, b, (short)0, c0, false, false);
        c1 = __builtin_amdgcn_wmma_f32_16x16x4_f32(false, a1, false, b, (short)0, c1, false, false);
        c2 = __builtin_amdgcn_wmma_f32_16x16x4_f32(false, a2, false, b, (short)0, c2, false, false);
        c3 = __builtin_amdgcn_wmma_f32_16x16x4_f32(false, a3, false, b, (short)0, c3, false, false);
    }
    float bv = bias[col];
    #pragma unroll
    for (int v = 0; v < 8; ++v) {
        int row = rm * 64 + 8 * half + v;
        __builtin_nontemporal_store(c0[v] + bv, &xz[(size_t)(row +  0) * G4 + col]);
        __builtin_nontemporal_store(c1[v] + bv, &xz[(size_t)(row + 16) * G4 + col]);
        __builtin_nontemporal_store(c2[v] + bv, &xz[(size_t)(row + 32) * G4 + col]);
        __builtin_nontemporal_store(c3[v] + bv, &xz[(size_t)(row + 48) * G4 + col]);
    }
}

// ---------------- Phase 2: persistent recurrence, R resident in LDS --------------------
template <bool HAS_XZ>
__global__ void __launch_bounds__(NTHREADS, 1)
lstm_persistent(const float* __restrict__ x, const float* __restrict__ W,
                const float* __restrict__ R, const float* __restrict__ bias,
                const float* __restrict__ wf, const float* __restrict__ wi,
                const float* __restrict__ wo, const float* __restrict__ xz,
                float* __restrict__ out, float* __restrict__ cstate,
                unsigned* __restrict__ ctr)
{
    extern __shared__ float smem[];
    float* Rt = smem;                       // [64 local cols][RT_STRIDE]  (transposed R)
    float* zb = smem + 64 * RT_STRIDE;      // [2 khalf][64 b][65]

    const int tid   = threadIdx.x;
    const int lane  = tid & 31;             // wave32
    const int wave  = tid >> 5;             // 0..31
    const int khalf = wave >> 4;            // K partition: 0 or 1
    const int w     = wave & 15;
    const int r     = w & 3;                // row tile: b in [16r,16r+16)
    const int g     = w >> 2;               // gate
    const int half  = lane >> 4;
    const int n     = lane & 15;
    const int cb    = blockIdx.x * 16;      // u slice base
    const int colg  = g * UNITS + cb + n;   // z-space column
    const int lc    = g * 16 + n;           // local column 0..63
    const int kbeg  = khalf * (UNITS / 2);

    // ---- stage R transposed into LDS (once) ----
    {
        const int lcol = tid & 63;
        const int kgrp = tid >> 6;          // 0..15
        const int gcol = (lcol >> 4) * UNITS + cb + (lcol & 15);
        for (int kk = 0; kk < 64; ++kk) {
            int k = kgrp + 16 * kk;
            Rt[lcol * RT_STRIDE + k] = R[(size_t)k * G4 + gcol];
        }
    }
    __syncthreads();

    for (int t = 0; t < T_STEPS; ++t) {
        v8f a0 = {}, a1 = {}, a2 = {}, a3 = {};
        // only khalf==0 carries bias / xz, so the LDS reduction is exact
        if (khalf == 0) {
            if (HAS_XZ) {
                #pragma unroll
                for (int v = 0; v < 8; ++v)
                    a0[v] = __builtin_nontemporal_load(
                        &xz[(size_t)(t * BATCH + 16 * r + 8 * half + v) * G4 + colg]);
            } else {
                float bv = bias[colg];
                #pragma unroll
                for (int v = 0; v < 8; ++v) a0[v] = bv;
            }
        }

        if (!HAS_XZ) {  // fallback: fused x@W from global (K split by khalf)
            const float* arow = x + ((size_t)(t * BATCH + 16 * r + n)) * DIM;
            const float* bp   = W + colg;
            for (int k = kbeg; k < kbeg + DIM / 2; k += 16) {
                #pragma unroll
                for (int j = 0; j < 4; ++j) {
                    int k4 = k + 4 * j;
                    v2f a = *(const v2f*)(arow + k4 + 2 * half);
                    v2f b;
                    b.x = bp[(size_t)(k4 + 2 * half)     * G4];
                    b.y = bp[(size_t)(k4 + 2 * half + 1) * G4];
                    v8f* accp = (j == 0) ? &a0 : (j == 1) ? &a1 : (j == 2) ? &a2 : &a3;
                    *accp = __builtin_amdgcn_wmma_f32_16x16x4_f32(
                                false, a, false, b, (short)0, *accp, false, false);
                }
            }
        }

        if (t > 0) {    // z += h_{t-1} @ R, B-fragments from LDS
            const float* hrow = out + ((size_t)((t - 1) * BATCH + 16 * r + n)) * UNITS;
            const float* rcol = Rt + lc * RT_STRIDE;
            for (int k = kbeg; k < kbeg + UNITS / 2; k += 16) {
                #pragma unroll
                for (int j = 0; j < 4; ++j) {
                    int k4 = k + 4 * j;
                    v2f a = *(const v2f*)(hrow + k4 + 2 * half);
                    v2f b = *(const v2f*)(rcol + k4 + 2 * half);   // ds_load_b64
                    v8f* accp = (j == 0) ? &a0 : (j == 1) ? &a1 : (j == 2) ? &a2 : &a3;
                    *accp = __builtin_amdgcn_wmma_f32_16x16x4_f32(
                                false, a, false, b, (short)0, *accp, false, false);
                }
            }
        }

        // ---- stash partial z tiles ----
        #pragma unroll
        for (int v = 0; v < 8; ++v)
            zb[(khalf * 64 + 16 * r + 8 * half + v) * 65 + 16 * g + n] =
                a0[v] + a1[v] + a2[v] + a3[v];
        __syncthreads();

        // ---- fused peephole gates: 1024 (b,u) items, 1 per thread ----
        {
            int b_ = tid >> 4;
            int ul = tid & 15;
            int ug = cb + ul;
            float z0 = zb[b_ * 65 + ul]      + zb[(64 + b_) * 65 + ul];
            float z1 = zb[b_ * 65 + 16 + ul] + zb[(64 + b_) * 65 + 16 + ul];
            float z2 = zb[b_ * 65 + 32 + ul] + zb[(64 + b_) * 65 + 32 + ul];
            float z3 = zb[b_ * 65 + 48 + ul] + zb[(64 + b_) * 65 + 48 + ul];
            float c  = cstate[b_ * UNITS + ug];
            float f  = sigf(z0 + wf[ug] * c);
            float ig = sigf(z1 + wi[ug] * c);
            float o  = sigf(z3 + wo[ug] * c);
            float cn = f * c + ig * tanhf(z2);
            float hn = o * tanhf(cn);
            cstate[b_ * UNITS + ug] = cn;
            out[((size_t)(t * BATCH) + b_) * UNITS + ug] = hn;
        }

        // prefetch next step's xz rows while waiting at the barrier
        if (HAS_XZ && khalf == 0 && t + 1 < T_STEPS) {
            #pragma unroll
            for (int v = 0; v < 8; ++v)
                __builtin_prefetch(
                    &xz[(size_t)((t + 1) * BATCH + 16 * r + 8 * half + v) * G4 + colg], 0, 1);
        }

        // ---- inter-step grid barrier (monotonic counter; relaxed-load spin) ----
        __threadfence();
        __syncthreads();
        if (tid == 0) {
            atomicAdd(ctr, 1u);
            unsigned target = (unsigned)(NBLOCKS * (t + 1));
            while (__hip_atomic_load(ctr, __ATOMIC_RELAXED, __HIP_MEMORY_SCOPE_AGENT) < target) {
                __builtin_amdgcn_s_sleep(8);
            }
        }
        __syncthreads();
        __threadfence();
    }

    // ---- final states: d_out layout = outputs | c_f | h_f ----
    {
        size_t off = (size_t)T_STEPS * BATCH * UNITS;
        int b_ = tid >> 4;
        int ul = tid & 15;
        int ug = cb + ul;
        out[off + (size_t)b_ * UNITS + ug] = cstate[b_ * UNITS + ug];
        out[off + (size_t)BATCH * UNITS + (size_t)b_ * UNITS + ug] =
            out[((size_t)((T_STEPS - 1) * BATCH) + b_) * UNITS + ug];
    }
}

extern "C" void kernel_launch(void* const* d_in, const int* in_sizes, int n_in,
                              void* d_out, int out_size, void* d_ws, size_t ws_size,
                              hipStream_t stream) {
    const float* x    = (const float*)d_in[0];
    const float* W    = (const float*)d_in[1];
    const float* R    = (const float*)d_in[2];
    const float* bias = (const float*)d_in[3];
    const float* wf   = (const float*)d_in[4];
    const float* wi   = (const float*)d_in[5];
    const float* wo   = (const float*)d_in[6];
    float* out = (float*)d_out;

    unsigned* ctr   = (unsigned*)d_ws;
    float*    cstat = (float*)((char*)d_ws + 256);
    float*    xz    = (float*)((char*)d_ws + (512 * 1024));
    const size_t xz_bytes = (size_t)T_STEPS * BATCH * G4 * sizeof(float);
    const bool   has_xz   = ws_size >= (512 * 1024 + xz_bytes);
    const size_t smem_b   = (size_t)SMEM_FLOATS * sizeof(float);

    lstm_init<<<64, 1024, 0, stream>>>(cstat, ctr);
    if (has_xz) {
        xz_gemm<<<16384, 256, 0, stream>>>(x, W, bias, xz);
        lstm_persistent<true><<<NBLOCKS, NTHREADS, smem_b, stream>>>(
            x, W, R, bias, wf, wi, wo, xz, out, cstat, ctr);
    } else {
        lstm_persistent<false><<<NBLOCKS, NTHREADS, smem_b, stream>>>(
            x, W, R, bias, wf, wi, wo, xz, out, cstat, ctr);
    }
}